// GraphEncoder_40870908789268
// MI455X (gfx1250) — compile-verified
//
#include <hip/hip_runtime.h>
#include <hip/hip_bf16.h>

#define N_NODES 50000
#define N_EDGES 800000
#define D 64
#define BN_EPS 1e-5f

typedef __attribute__((ext_vector_type(2))) float v2f;
typedef __attribute__((ext_vector_type(8))) float v8f;

// ---------------------------------------------------------------------------
// xa[node] = xb[node] = emb[x_ids[node]]  (dual write: x and h-init for agg)
// ---------------------------------------------------------------------------
__global__ void k_gather(const int* __restrict__ ids,
                         const float* __restrict__ emb,
                         float* __restrict__ xa,
                         float* __restrict__ xb) {
    int t = blockIdx.x * blockDim.x + threadIdx.x;   // N_NODES*16 threads
    if (t >= N_NODES * 16) return;
    int node = t >> 4;
    int c    = (t & 15) * 4;
    int id   = ids[node];
    float4 v = *(const float4*)(emb + (size_t)id * D + c);
    *(float4*)(xa + (size_t)node * D + c) = v;
    *(float4*)(xb + (size_t)node * D + c) = v;
}

// ---------------------------------------------------------------------------
// h[dst] += x[src] for each edge.  One thread per (edge, 4-feature chunk).
// x and h (12.8MB each) are L2-resident -> gathers and atomics stay on-chip.
// ---------------------------------------------------------------------------
__global__ void k_aggregate(const int* __restrict__ ei,
                            const float* __restrict__ x,
                            float* __restrict__ h) {
    long long t = (long long)blockIdx.x * blockDim.x + threadIdx.x;
    if (t >= (long long)N_EDGES * 16) return;
    int e = (int)(t >> 4);
    int c = ((int)t & 15) * 4;
    int s = ei[e];
    int d = ei[N_EDGES + e];
    float4 v = *(const float4*)(x + (size_t)s * D + c);
    float* hp = h + (size_t)d * D + c;
    atomicAdd(hp + 0, v.x);
    atomicAdd(hp + 1, v.y);
    atomicAdd(hp + 2, v.z);
    atomicAdd(hp + 3, v.w);
}

// ---------------------------------------------------------------------------
// Z = A @ W + bias, fused per-column sum / sum-of-squares for BatchNorm.
// Block = 128 threads = 4 waves; block handles 16 rows x 64 cols.
// W (16KB) and the 16x64 A tile (4KB) are staged in LDS; the K-loop feeds
// V_WMMA_F32_16X16X4_F32 (fp32 WMMA) entirely from ds_load.
// 50000 % 16 == 0 -> no predication; EXEC all-ones for WMMA.
// ---------------------------------------------------------------------------
__global__ void __launch_bounds__(128)
k_gemm_stats(const float* __restrict__ A,     // [N_NODES, 64]
             const float* __restrict__ W,     // [64, 64] row-major
             const float* __restrict__ bias,  // [64]
             float* __restrict__ Z,           // [N_NODES, 64]
             float* __restrict__ colsum,      // [64]
             float* __restrict__ colsumsq) {  // [64]
    __shared__ float lw[D * D];   // 16 KB
    __shared__ float la[16 * D];  //  4 KB
    const int tid  = threadIdx.x;
    const int wave = tid >> 5;    // 0..3 -> N tile
    const int lane = tid & 31;
    const int lh   = lane >> 4;   // lane half (0/1)
    const int lm   = lane & 15;
    const int m0   = blockIdx.x * 16;
    const int n0   = wave * 16;

    // cooperative stage: W = 1024 float4 (8/thread), A tile = 256 float4 (2/thread)
    {
        const float4* ws = (const float4*)W;
        float4*       wd = (float4*)lw;
#pragma unroll
        for (int i = 0; i < 8; ++i) wd[tid + i * 128] = ws[tid + i * 128];
        const float4* as = (const float4*)(A + (size_t)m0 * D);
        float4*       ad = (float4*)la;
#pragma unroll
        for (int i = 0; i < 2; ++i) ad[tid + i * 128] = as[tid + i * 128];
    }
    __syncthreads();

    v8f c = {};
#pragma unroll
    for (int kk = 0; kk < 16; ++kk) {
        const int ka = kk * 4 + lh * 2;
        v2f a;                                   // A[m0+lm][ka..ka+1]
        a.x = la[lm * D + ka];
        a.y = la[lm * D + ka + 1];
        v2f b;                                   // W[ka..ka+1][n0+lm]
        b.x = lw[ka * D + n0 + lm];
        b.y = lw[(ka + 1) * D + n0 + lm];
        c = __builtin_amdgcn_wmma_f32_16x16x4_f32(
                /*neg_a=*/false, a, /*neg_b=*/false, b,
                /*c_mod=*/(short)0, c, /*reuse_a=*/false, /*reuse_b=*/false);
    }

    // C layout: VGPR i, lane half lh -> row m0 + i + lh*8, col n0 + lm
    const float bv = bias[n0 + lm];
    float s = 0.f, q = 0.f;
    float* zp = Z + (size_t)(m0 + lh * 8) * D + n0 + lm;
#pragma unroll
    for (int i = 0; i < 8; ++i) {
        float v = c[i] + bv;
        zp[(size_t)i * D] = v;
        s += v;
        q += v * v;
    }
    // combine the two row-halves of this column (lane L <-> L^16)
    s += __shfl_xor(s, 16, 32);
    q += __shfl_xor(q, 16, 32);
    if (lh == 0) {
        atomicAdd(&colsum[n0 + lm],   s);
        atomicAdd(&colsumsq[n0 + lm], q);
    }
}

// ---------------------------------------------------------------------------
// Zero BN accumulators + pooled (320 floats).
// ---------------------------------------------------------------------------
__global__ void k_zero_stats(float* __restrict__ stats) {
    int t = threadIdx.x;
    if (t < 320) stats[t] = 0.f;
}

// ---------------------------------------------------------------------------
// scale = g * rsqrt(var + eps);  shift = be - mean*scale  (biased variance)
// ---------------------------------------------------------------------------
__global__ void k_bn_finalize(const float* __restrict__ colsum,
                              const float* __restrict__ colsumsq,
                              const float* __restrict__ g,
                              const float* __restrict__ be,
                              float* __restrict__ scale,
                              float* __restrict__ shift) {
    int j = threadIdx.x;
    if (j < D) {
        const float inv_n = 1.0f / (float)N_NODES;
        float m  = colsum[j] * inv_n;
        float v  = colsumsq[j] * inv_n - m * m;
        float sc = g[j] * rsqrtf(v + BN_EPS);
        scale[j] = sc;
        shift[j] = be[j] - m * sc;
    }
}

// ---------------------------------------------------------------------------
// y = relu(z*scale + shift), one float4/thread (grid sized exactly).
// Optional dual write (y2 = h-init for the next layer's aggregation) and
// optional fused global_add_pool through LDS ds_add_f32.
// ---------------------------------------------------------------------------
__global__ void k_bn_relu(const float4* __restrict__ z,
                          const float* __restrict__ scale,
                          const float* __restrict__ shift,
                          float4* __restrict__ y,
                          float4* __restrict__ y2,      // may be nullptr
                          float* __restrict__ pooled,
                          int do_pool) {
    __shared__ float cs[D];
    int t = blockIdx.x * blockDim.x + threadIdx.x;   // 800000 float4, exact
    if (threadIdx.x < D) cs[threadIdx.x] = 0.f;
    __syncthreads();

    const int c0 = (t & 15) * 4;
    float4 v = z[t];
    v.x = fmaxf(v.x * scale[c0 + 0] + shift[c0 + 0], 0.f);
    v.y = fmaxf(v.y * scale[c0 + 1] + shift[c0 + 1], 0.f);
    v.z = fmaxf(v.z * scale[c0 + 2] + shift[c0 + 2], 0.f);
    v.w = fmaxf(v.w * scale[c0 + 3] + shift[c0 + 3], 0.f);
    y[t] = v;
    if (y2) y2[t] = v;

    if (do_pool) {
        atomicAdd(&cs[c0 + 0], v.x);   // ds_add_f32
        atomicAdd(&cs[c0 + 1], v.y);
        atomicAdd(&cs[c0 + 2], v.z);
        atomicAdd(&cs[c0 + 3], v.w);
        __syncthreads();
        if (threadIdx.x < D) atomicAdd(&pooled[threadIdx.x], cs[threadIdx.x]);
    }
}

// ---------------------------------------------------------------------------
// out = pooled @ lin_w + lin_b   (1x64 @ 64x64)
// ---------------------------------------------------------------------------
__global__ void k_final_linear(const float* __restrict__ pooled,
                               const float* __restrict__ w,
                               const float* __restrict__ b,
                               float* __restrict__ out) {
    int j = threadIdx.x;
    if (j < D) {
        float acc = b[j];
#pragma unroll 8
        for (int k = 0; k < D; ++k) acc += pooled[k] * w[k * D + j];
        out[j] = acc;
    }
}

// ---------------------------------------------------------------------------
extern "C" void kernel_launch(void* const* d_in, const int* in_sizes, int n_in,
                              void* d_out, int out_size, void* d_ws, size_t ws_size,
                              hipStream_t stream) {
    // inputs (setup_inputs order)
    const int*   x_ids = (const int*)  d_in[0];
    const int*   ei    = (const int*)  d_in[1];
    const float* emb   = (const float*)d_in[2];
    const float* c1_w1 = (const float*)d_in[3],  *c1_b1 = (const float*)d_in[4];
    const float* c1_g1 = (const float*)d_in[5],  *c1_be1= (const float*)d_in[6];
    const float* c1_w2 = (const float*)d_in[7],  *c1_b2 = (const float*)d_in[8];
    const float* c1_g2 = (const float*)d_in[9],  *c1_be2= (const float*)d_in[10];
    const float* c2_w1 = (const float*)d_in[11], *c2_b1 = (const float*)d_in[12];
    const float* c2_g1 = (const float*)d_in[13], *c2_be1= (const float*)d_in[14];
    const float* c2_w2 = (const float*)d_in[15], *c2_b2 = (const float*)d_in[16];
    const float* c2_g2 = (const float*)d_in[17], *c2_be2= (const float*)d_in[18];
    const float* lin_w = (const float*)d_in[19], *lin_b = (const float*)d_in[20];
    float* out = (float*)d_out;

    // workspace layout (floats): A | B | C | stats(320)
    const size_t NF = (size_t)N_NODES * D;
    float* A = (float*)d_ws;
    float* B = A + NF;
    float* C = B + NF;
    float* stats  = C + NF;
    float* sum    = stats;
    float* sumsq  = stats + 64;
    float* scale  = stats + 128;
    float* shift  = stats + 192;
    float* pooled = stats + 256;

    const int TB = 256;
    const int g_gather = (N_NODES * 16 + TB - 1) / TB;                    // 3125
    const int g_agg    = (int)(((long long)N_EDGES * 16 + TB - 1) / TB);  // 50000
    const int g_gemm   = N_NODES / 16;                                    // 3125 (exact)
    const int g_apply  = (N_NODES * 16) / TB;                             // 3125 (exact)

    // x -> A, h-init -> B
    k_gather<<<g_gather, TB, 0, stream>>>(x_ids, emb, A, B);

    // ===================== GIN layer 1 =====================
    k_aggregate<<<g_agg, TB, 0, stream>>>(ei, A, B);               // B = x + sum_nbr x

    k_zero_stats<<<1, 320, 0, stream>>>(stats);
    k_gemm_stats<<<g_gemm, 128, 0, stream>>>(B, c1_w1, c1_b1, C, sum, sumsq);
    k_bn_finalize<<<1, 64, 0, stream>>>(sum, sumsq, c1_g1, c1_be1, scale, shift);
    k_bn_relu<<<g_apply, TB, 0, stream>>>((float4*)C, scale, shift,
                                          (float4*)C, nullptr, pooled, 0);

    k_zero_stats<<<1, 320, 0, stream>>>(stats);
    k_gemm_stats<<<g_gemm, 128, 0, stream>>>(C, c1_w2, c1_b2, A, sum, sumsq);
    k_bn_finalize<<<1, 64, 0, stream>>>(sum, sumsq, c1_g2, c1_be2, scale, shift);
    // x1 -> A, and dual-write h-init for layer 2 -> B
    k_bn_relu<<<g_apply, TB, 0, stream>>>((float4*)A, scale, shift,
                                          (float4*)A, (float4*)B, pooled, 0);

    // ===================== GIN layer 2 =====================
    k_aggregate<<<g_agg, TB, 0, stream>>>(ei, A, B);               // B = x1 + sum_nbr x1

    k_zero_stats<<<1, 320, 0, stream>>>(stats);
    k_gemm_stats<<<g_gemm, 128, 0, stream>>>(B, c2_w1, c2_b1, C, sum, sumsq);
    k_bn_finalize<<<1, 64, 0, stream>>>(sum, sumsq, c2_g1, c2_be1, scale, shift);
    k_bn_relu<<<g_apply, TB, 0, stream>>>((float4*)C, scale, shift,
                                          (float4*)C, nullptr, pooled, 0);

    k_zero_stats<<<1, 320, 0, stream>>>(stats);                    // also zeroes pooled
    k_gemm_stats<<<g_gemm, 128, 0, stream>>>(C, c2_w2, c2_b2, B, sum, sumsq);
    k_bn_finalize<<<1, 64, 0, stream>>>(sum, sumsq, c2_g2, c2_be2, scale, shift);
    k_bn_relu<<<g_apply, TB, 0, stream>>>((float4*)B, scale, shift,
                                          (float4*)B, nullptr, pooled, 1);  // fused pool

    // out = pooled @ lin_w + lin_b
    k_final_linear<<<1, 64, 0, stream>>>(pooled, lin_w, lin_b, out);
    (void)n_in; (void)in_sizes; (void)out_size; (void)ws_size;
}